// StrategicGNN_26980984553998
// MI455X (gfx1250) — compile-verified
//
#include <hip/hip_runtime.h>
#include <hip/hip_bf16.h>
#include <math.h>

typedef __attribute__((ext_vector_type(2))) float v2f;
typedef __attribute__((ext_vector_type(8))) float v8f;

// ---------------------------------------------------------------------------
// helpers
// ---------------------------------------------------------------------------
__device__ inline float wave_sum32(float v) {
#pragma unroll
  for (int off = 16; off > 0; off >>= 1) v += __shfl_xor(v, off, 32);
  return v;
}

// float atomic max via signed-max / unsigned-min trick (init value must be -inf)
__device__ inline void atomic_max_f32(float* addr, float val) {
  if (val >= 0.0f) atomicMax((int*)addr, __float_as_int(val));
  else             atomicMin((unsigned int*)addr, __float_as_uint(val));
}

__global__ void fill_kernel(float* __restrict__ p, float v, long n) {
  long i      = (long)blockIdx.x * blockDim.x + threadIdx.x;
  long stride = (long)gridDim.x * blockDim.x;
  for (; i < n; i += stride) p[i] = v;
}

// ---------------------------------------------------------------------------
// WMMA fp32 GEMM:  Y[M,F] = act( X[M,K] * W[F,K]^T + bias )
// One wave computes a 32(M) x 64(N) tile: 2 A-frags x 4 B-frags = 8 v8f accs.
// K-step 4, software-pipelined: next step's 6 float2 loads are issued before
// the current step's 8 back-to-back WMMAs (C-accumulate chain needs no NOPs).
// A 16x4 f32 fragment: lane(l): row = l%16, holds K = {2*(l/16), 2*(l/16)+1}
// B 4x16 f32 fragment: lane(l): col = l%16, holds same K pair
// D 16x16 f32: vgpr v, lane l -> (m = v + 8*(l/16), n = l%16)
// ---------------------------------------------------------------------------
__global__ void wmma_gemm_f32(const float* __restrict__ X, int ldx,
                              const float* __restrict__ W,
                              const float* __restrict__ bias,
                              float* __restrict__ Y, int ldy,
                              int M, int K, int F, int do_relu) {
  const int lane = threadIdx.x & 31;
  const int wid  = (int)((blockIdx.x * blockDim.x + threadIdx.x) >> 5);
  const int tiles_n = F >> 6;   // F multiple of 64
  const int tiles_m = M >> 5;   // M multiple of 32
  if (wid >= tiles_m * tiles_n) return;
  const int tm = wid / tiles_n;
  const int tn = wid % tiles_n;

  const int lo = lane & 15;
  const int hi = lane >> 4;

  const float* xr_[2];
  xr_[0] = X + (size_t)(tm * 32 + lo) * ldx + 2 * hi;
  xr_[1] = xr_[0] + (size_t)16 * ldx;
  const float* wr_[4];
#pragma unroll
  for (int t = 0; t < 4; ++t)
    wr_[t] = W + (size_t)(tn * 64 + t * 16 + lo) * K + 2 * hi;

  v8f acc[2][4];
#pragma unroll
  for (int f = 0; f < 2; ++f)
#pragma unroll
    for (int t = 0; t < 4; ++t) acc[f][t] = (v8f){};

  v2f a[2], b[4], na[2], nb[4];
#pragma unroll
  for (int f = 0; f < 2; ++f) a[f] = *(const v2f*)(xr_[f]);
#pragma unroll
  for (int t = 0; t < 4; ++t) b[t] = *(const v2f*)(wr_[t]);

  for (int k = 4; k < K; k += 4) {
    // prefetch next K-step while current WMMAs execute
#pragma unroll
    for (int f = 0; f < 2; ++f) na[f] = *(const v2f*)(xr_[f] + k);
#pragma unroll
    for (int t = 0; t < 4; ++t) nb[t] = *(const v2f*)(wr_[t] + k);
#pragma unroll
    for (int f = 0; f < 2; ++f)
#pragma unroll
      for (int t = 0; t < 4; ++t)
        acc[f][t] = __builtin_amdgcn_wmma_f32_16x16x4_f32(
            false, a[f], false, b[t], (short)0, acc[f][t], false, false);
#pragma unroll
    for (int f = 0; f < 2; ++f) a[f] = na[f];
#pragma unroll
    for (int t = 0; t < 4; ++t) b[t] = nb[t];
  }
  // tail step
#pragma unroll
  for (int f = 0; f < 2; ++f)
#pragma unroll
    for (int t = 0; t < 4; ++t)
      acc[f][t] = __builtin_amdgcn_wmma_f32_16x16x4_f32(
          false, a[f], false, b[t], (short)0, acc[f][t], false, false);

#pragma unroll
  for (int f = 0; f < 2; ++f) {
#pragma unroll
    for (int v = 0; v < 8; ++v) {
      const int m = tm * 32 + f * 16 + v + 8 * hi;
#pragma unroll
      for (int t = 0; t < 4; ++t) {
        const int n = tn * 64 + t * 16 + lo;
        float x = acc[f][t][v];
        if (bias) x += bias[n];
        if (do_relu) x = fmaxf(x, 0.0f);
        Y[(size_t)m * ldy + n] = x;
      }
    }
  }
}

// ---------------------------------------------------------------------------
// GATv2 pass A: per-edge attention logits + per-(dst,head) running max.
// One wave per edge (self-loop edges appended: e in [nE, nE+nN) -> src=dst=e-nE).
// ---------------------------------------------------------------------------
__global__ void gat_logits_kernel(const float* __restrict__ xl,
                                  const float* __restrict__ xr,
                                  const int* __restrict__ ei,
                                  const float* __restrict__ att,
                                  float* __restrict__ logits,
                                  float* __restrict__ mmax,
                                  int nE, int nN, int H, int C) {
  const int wid  = (int)((blockIdx.x * blockDim.x + threadIdx.x) >> 5);
  const int lane = threadIdx.x & 31;
  if (wid >= nE + nN) return;
  int s, d;
  if (wid < nE) { s = ei[wid]; d = ei[nE + wid]; }
  else          { s = d = wid - nE; }

  const int D = H * C;
  float part[4] = {0.f, 0.f, 0.f, 0.f};
  for (int c = lane; c < D; c += 32) {
    float e = xl[(size_t)s * D + c] + xr[(size_t)d * D + c];
    e = (e > 0.0f) ? e : 0.2f * e;            // leaky_relu(0.2)
    part[c / C] += e * att[c];                // att is [H,C] row-major
  }
  for (int h = 0; h < H; ++h) {
    float v = wave_sum32(part[h]);
    if (lane == 0) {
      logits[(size_t)wid * H + h] = v;
      atomic_max_f32(&mmax[(size_t)d * H + h], v);
    }
  }
}

// ---------------------------------------------------------------------------
// GATv2 pass B: p = exp(logit - max[dst]); denom[dst] += p; agg[dst] += p*xl[src]
// ---------------------------------------------------------------------------
__global__ void gat_scatter_kernel(const float* __restrict__ xl,
                                   const int* __restrict__ ei,
                                   const float* __restrict__ logits,
                                   const float* __restrict__ mmax,
                                   float* __restrict__ denom,
                                   float* __restrict__ agg,
                                   int nE, int nN, int H, int C) {
  const int wid  = (int)((blockIdx.x * blockDim.x + threadIdx.x) >> 5);
  const int lane = threadIdx.x & 31;
  if (wid >= nE + nN) return;
  int s, d;
  if (wid < nE) { s = ei[wid]; d = ei[nE + wid]; }
  else          { s = d = wid - nE; }

  const int D = H * C;
  float p[4];
  for (int h = 0; h < H; ++h)
    p[h] = expf(logits[(size_t)wid * H + h] - mmax[(size_t)d * H + h]);
  if (lane == 0)
    for (int h = 0; h < H; ++h) atomicAdd(&denom[(size_t)d * H + h], p[h]);
  for (int c = lane; c < D; c += 32)
    atomicAdd(&agg[(size_t)d * D + c], p[c / C] * xl[(size_t)s * D + c]);
}

// ---------------------------------------------------------------------------
// GATv2 pass C: out = act( agg / denom + bias )
// ---------------------------------------------------------------------------
__global__ void gat_finalize_kernel(const float* __restrict__ agg,
                                    const float* __restrict__ denom,
                                    const float* __restrict__ bias,
                                    float* __restrict__ out,
                                    int nN, int H, int C, int do_relu) {
  const int D   = H * C;
  const long n  = (long)nN * D;
  long idx = (long)blockIdx.x * blockDim.x + threadIdx.x;
  if (idx >= n) return;
  const int node = (int)(idx / D);
  const int c    = (int)(idx % D);
  float v = agg[idx] / denom[(size_t)node * H + c / C] + bias[c];
  if (do_relu) v = fmaxf(v, 0.0f);
  out[idx] = v;
}

// ---------------------------------------------------------------------------
// LayerNorm over D=256, one wave per node (8 values per lane)
// ---------------------------------------------------------------------------
__global__ void layernorm_kernel(const float* __restrict__ h,
                                 const float* __restrict__ g,
                                 const float* __restrict__ b,
                                 float* __restrict__ out, int nN) {
  const int wid  = (int)((blockIdx.x * blockDim.x + threadIdx.x) >> 5);
  const int lane = threadIdx.x & 31;
  if (wid >= nN) return;
  const float* row = h + (size_t)wid * 256;
  float vals[8];
  float s = 0.f;
#pragma unroll
  for (int i = 0; i < 8; ++i) { vals[i] = row[lane + 32 * i]; s += vals[i]; }
  const float mu = wave_sum32(s) * (1.0f / 256.0f);
  float vs = 0.f;
#pragma unroll
  for (int i = 0; i < 8; ++i) { float dv = vals[i] - mu; vs += dv * dv; }
  const float var  = wave_sum32(vs) * (1.0f / 256.0f);
  const float rstd = rsqrtf(var + 1e-5f);
#pragma unroll
  for (int i = 0; i < 8; ++i) {
    const int c = lane + 32 * i;
    out[(size_t)wid * 256 + c] = (vals[i] - mu) * rstd * g[c] + b[c];
  }
}

// ---------------------------------------------------------------------------
// host orchestration
// ---------------------------------------------------------------------------
extern "C" void kernel_launch(void* const* d_in, const int* in_sizes, int n_in,
                              void* d_out, int out_size, void* d_ws, size_t ws_size,
                              hipStream_t stream) {
  const int N = in_sizes[0] / 128;   // NODE_F = 128
  const int E = in_sizes[2] / 2;

  const float* node_f = (const float*)d_in[0];
  const float* msg_f  = (const float*)d_in[1];
  const int*   ei     = (const int*)d_in[2];
  const float* me_w1  = (const float*)d_in[3];  const float* me_b1 = (const float*)d_in[4];
  const float* me_w2  = (const float*)d_in[5];  const float* me_b2 = (const float*)d_in[6];
  const float* ne_w1  = (const float*)d_in[7];  const float* ne_b1 = (const float*)d_in[8];
  const float* ne_w2  = (const float*)d_in[9];  const float* ne_b2 = (const float*)d_in[10];
  const float* c1_wl  = (const float*)d_in[11]; const float* c1_wr = (const float*)d_in[12];
  const float* c1_att = (const float*)d_in[13]; const float* c1_b  = (const float*)d_in[14];
  const float* c2_wl  = (const float*)d_in[15]; const float* c2_wr = (const float*)d_in[16];
  const float* c2_att = (const float*)d_in[17]; const float* c2_b  = (const float*)d_in[18];
  const float* c3_wl  = (const float*)d_in[19]; const float* c3_wr = (const float*)d_in[20];
  const float* c3_att = (const float*)d_in[21]; const float* c3_b  = (const float*)d_in[22];
  const float* en_w1  = (const float*)d_in[23]; const float* en_b1 = (const float*)d_in[24];
  const float* en_w2  = (const float*)d_in[25]; const float* en_b2 = (const float*)d_in[26];
  const float* ln_g   = (const float*)d_in[27]; const float* ln_b  = (const float*)d_in[28];
  float* out = (float*)d_out;

  // workspace bump allocator
  float* ws = (float*)d_ws;
  size_t off = 0;
  auto alloc = [&](size_t nfl) { float* p = ws + off; off += nfl; return p; };
  float* X   = alloc((size_t)N * 256);        // node features between layers
  float* TMP = alloc((size_t)N * 256);        // MLP hidden
  float* XL  = alloc((size_t)N * 256);        // GAT source transform
  float* XR  = alloc((size_t)N * 256);        // GAT target transform (contig after XL)
  float* AGG = alloc((size_t)N * 256);        // aggregation / pre-LN buffer
  float* LOG = alloc((size_t)(E + N) * 4);    // per-edge logits
  float* MM  = alloc((size_t)N * 4);          // segment max
  float* DEN = alloc((size_t)N * 4);          // softmax denom
  (void)ws_size; (void)n_in; (void)out_size;

  auto fill = [&](float* p, float v, long n) {
    int blocks = (int)((n + 255) / 256); if (blocks > 2048) blocks = 2048;
    fill_kernel<<<blocks, 256, 0, stream>>>(p, v, n);
  };
  auto gemm = [&](const float* Xp, int ldx, const float* Wp, const float* bp,
                  float* Yp, int ldy, int M, int K, int F, int relu) {
    int waves  = (M >> 5) * (F >> 6);
    int blocks = (waves * 32 + 255) / 256;
    wmma_gemm_f32<<<blocks, 256, 0, stream>>>(Xp, ldx, Wp, bp, Yp, ldy, M, K, F, relu);
  };
  auto conv = [&](const float* xin, int Kin, const float* wl, const float* wr,
                  const float* att, const float* bias, int H, int C, int relu_out,
                  float* xout) {
    const int D = H * C;
    gemm(xin, Kin, wl, nullptr, XL, D, N, Kin, D, 0);
    gemm(xin, Kin, wr, nullptr, XR, D, N, Kin, D, 0);
    fill(MM, -__builtin_huge_valf(), (long)N * H);
    fill(DEN, 0.0f, (long)N * H);
    fill(AGG, 0.0f, (long)N * D);
    const int nW = E + N;
    const int eb = (nW * 32 + 255) / 256;
    gat_logits_kernel<<<eb, 256, 0, stream>>>(XL, XR, ei, att, LOG, MM, E, N, H, C);
    gat_scatter_kernel<<<eb, 256, 0, stream>>>(XL, ei, LOG, MM, DEN, AGG, E, N, H, C);
    const long nfin = (long)N * D;
    gat_finalize_kernel<<<(int)((nfin + 255) / 256), 256, 0, stream>>>(
        AGG, DEN, bias, xout, N, H, C, relu_out);
  };

  // encoders: msg 256->256(relu)->128 ; node 128->128(relu)->128 ; concat into X[N,256]
  gemm(msg_f, 256, me_w1, me_b1, TMP, 256, N, 256, 256, 1);
  gemm(TMP, 256, me_w2, me_b2, X + 128, 256, N, 256, 128, 0);
  gemm(node_f, 128, ne_w1, ne_b1, TMP, 128, N, 128, 128, 1);
  gemm(TMP, 128, ne_w2, ne_b2, X, 256, N, 128, 128, 0);

  // conv1: 256 -> 4x64 concat, relu     (X [N,256] -> X [N,256])
  conv(X, 256, c1_wl, c1_wr, c1_att, c1_b, 4, 64, 1, X);
  // conv2: 256 -> 2x64 concat, relu     (X [N,256] -> X [N,128] tight)
  conv(X, 256, c2_wl, c2_wr, c2_att, c2_b, 2, 64, 1, X);
  // conv3: 128 -> 1x256 mean(=identity) (X [N,128] -> X [N,256])
  conv(X, 128, c3_wl, c3_wr, c3_att, c3_b, 1, 256, 0, X);

  // enhancer: 256 -> 512 (relu) -> 256, then LayerNorm -> out
  gemm(X, 256, en_w1, en_b1, XL /*[N,512] spans XL+XR*/, 512, N, 256, 512, 1);
  gemm(XL, 512, en_w2, en_b2, AGG, 256, N, 512, 256, 0);
  layernorm_kernel<<<(N * 32 + 255) / 256, 256, 0, stream>>>(AGG, ln_g, ln_b, out, N);
}